// Conv2D_84748294685300
// MI455X (gfx1250) — compile-verified
//
#include <hip/hip_runtime.h>

// CDNA5 / gfx1250: wave32, WMMA (not MFMA).
// 3x3 VALID conv (4096x4096 fp32 -> 4094x4094 fp32) + bias.
// Memory-bound (~134 MB @ 23.3 TB/s ~= 5.8 us floor). Compute done exactly in
// fp32 via V_WMMA_F32_16X16X4_F32 using a Toeplitz (im2col-band) formulation:
//   D[m,n] = sum_{r=0..2} sum_{j=0..19} x[i0+m+r, j0+j] * W[r, j-n]
// 15 chained K=4 WMMAs per 16x16 output tile, accumulating in the f32 C/D frag.
// Staging: interior blocks use GLOBAL_LOAD_ASYNC_TO_LDS_B128 (ASYNCcnt) for
// 128-bit global->LDS copies with no VGPR round-trip; boundary blocks fall
// back to clamped scalar staging (keeps LDS finite => no NaN*0 contamination).

typedef __attribute__((ext_vector_type(2))) float v2f;
typedef __attribute__((ext_vector_type(8))) float v8f;

#define IMG_H 4096
#define IMG_W 4096
#define OUT_H 4094
#define OUT_W 4094

#define LDS_ROWS 18
#define LDS_COLS 132   // 128 block-output cols + 2 halo + 2 K-pad; 33 x 16B per row

#if defined(__has_builtin)
#if __has_builtin(__builtin_amdgcn_global_load_async_to_lds_b128) && \
    __has_builtin(__builtin_amdgcn_s_wait_asynccnt)
#define USE_ASYNC_LDS 1
#endif
#endif
#ifndef USE_ASYNC_LDS
#define USE_ASYNC_LDS 0
#endif

#if USE_ASYNC_LDS
// Builtin signature (from hipcc diagnostic): param0 is
// 'int __attribute__((vector_size(16))) __device__ *'  (AS1 = global);
// param1 expected AS3 (__shared__) by symmetry; params 2,3 are imm ints.
typedef int v4i __attribute__((vector_size(16)));
#define GLOBAL_AS __attribute__((address_space(1)))
#define LDS_AS    __attribute__((address_space(3)))
#endif

__global__ __launch_bounds__(256) void conv3x3_wmma_f32(
    const float* __restrict__ x, const float* __restrict__ wgt,
    const float* __restrict__ bias, float* __restrict__ y)
{
    __shared__ float tile[LDS_ROWS * LDS_COLS];

    const int tid  = threadIdx.x;
    const int wave = tid >> 5;     // 0..7 : which 16-col sub-tile
    const int lane = tid & 31;
    const int half = lane >> 4;    // 0 = lanes 0-15, 1 = lanes 16-31
    const int lo   = lane & 15;

    const int I0 = blockIdx.y * 16;    // output row base (block)
    const int C0 = blockIdx.x * 128;   // output col base (block)
    const int J0 = C0 + wave * 16;     // output col base (this wave)

    // ---- stage the 18x132 input tile into LDS ----
    const bool interior = (I0 + LDS_ROWS <= IMG_H) && (C0 + LDS_COLS <= IMG_W);
    if (interior) {
#if USE_ASYNC_LDS
        // 18 rows x 33 chunks of 16B; global and LDS addrs both 16B-aligned.
        for (int idx = tid; idx < LDS_ROWS * (LDS_COLS / 4); idx += 256) {
            int r  = idx / (LDS_COLS / 4);
            int c4 = idx - r * (LDS_COLS / 4);
            const float* g = x + (size_t)(I0 + r) * IMG_W + C0 + c4 * 4;
            float*       l = &tile[r * LDS_COLS + c4 * 4];
            __builtin_amdgcn_global_load_async_to_lds_b128(
                (GLOBAL_AS v4i*)g, (LDS_AS v4i*)l, /*offset=*/0, /*cpol=*/0);
        }
        __builtin_amdgcn_s_wait_asynccnt(0);
#else
        for (int idx = tid; idx < LDS_ROWS * (LDS_COLS / 4); idx += 256) {
            int r  = idx / (LDS_COLS / 4);
            int c4 = idx - r * (LDS_COLS / 4);
            const float4* g =
                (const float4*)(x + (size_t)(I0 + r) * IMG_W + C0 + c4 * 4);
            *(float4*)&tile[r * LDS_COLS + c4 * 4] = *g;
        }
#endif
    } else {
        // clamped scalar staging: always finite data in LDS, so padded-K WMMA
        // taps can never inject NaN (WMMA: any NaN input -> NaN output)
        for (int idx = tid; idx < LDS_ROWS * LDS_COLS; idx += 256) {
            int r  = idx / LDS_COLS;
            int c  = idx - r * LDS_COLS;
            int gr = I0 + r; if (gr > IMG_H - 1) gr = IMG_H - 1;
            int gc = C0 + c; if (gc > IMG_W - 1) gc = IMG_W - 1;
            tile[idx] = x[gr * IMG_W + gc];
        }
    }
    __syncthreads();

    // 3x3 weights + bias into registers (uniform, L2-resident)
    const float w00 = wgt[0], w01 = wgt[1], w02 = wgt[2];
    const float w10 = wgt[3], w11 = wgt[4], w12 = wgt[5];
    const float w20 = wgt[6], w21 = wgt[7], w22 = wgt[8];
    const float bns = bias[0];

    v8f acc = {};                       // 16x16 f32 C/D fragment (8 VGPRs)
    const v2f* lds2 = (const v2f*)tile; // A loads as float2 (ds_load_b64)

    const int k0 = half * 2;            // A/B frag: lanes 0-15 hold K={0,1}, lanes 16-31 K={2,3}

#pragma unroll
    for (int r = 0; r < 3; ++r) {
        const float wr0 = (r == 0) ? w00 : (r == 1) ? w10 : w20;
        const float wr1 = (r == 0) ? w01 : (r == 1) ? w11 : w21;
        const float wr2 = (r == 0) ? w02 : (r == 1) ? w12 : w22;
        // A-matrix row for this lane: m = lo, input row = m + r; local col base
        const int arow = (lo + r) * LDS_COLS + wave * 16;
#pragma unroll
        for (int c0 = 0; c0 < 20; c0 += 4) {
            // A[m, k] = x_tile[m + r, wave*16 + c0 + k]  (k = k0, k0+1)
            v2f a = lds2[(arow + c0 + k0) >> 1];   // index always even

            // B[k, n] = W[r, (c0+k) - n] inside the 3-tap band, else 0
            const int d0 = c0 + k0 - lo;
            const int d1 = d0 + 1;
            v2f bb;
            bb.x = (d0 == 0) ? wr0 : (d0 == 1) ? wr1 : (d0 == 2) ? wr2 : 0.0f;
            bb.y = (d1 == 0) ? wr0 : (d1 == 1) ? wr1 : (d1 == 2) ? wr2 : 0.0f;

            // D = A(16x4,f32) x B(4x16,f32) + C  -- exact fp32 path
            acc = __builtin_amdgcn_wmma_f32_16x16x4_f32(
                false, a, false, bb, (short)0, acc, false, false);
        }
    }

    // ---- store: C/D layout -> VGPR v holds rows {v, v+8}, N = lane&15 ----
    const int col = J0 + lo;
#pragma unroll
    for (int v = 0; v < 8; ++v) {
        const int row = I0 + v + half * 8;
        if (row < OUT_H && col < OUT_W)
            y[row * OUT_W + col] = acc[v] + bns;
    }
}

extern "C" void kernel_launch(void* const* d_in, const int* in_sizes, int n_in,
                              void* d_out, int out_size, void* d_ws, size_t ws_size,
                              hipStream_t stream) {
    const float* x    = (const float*)d_in[0];   // 4096*4096
    const float* wgt  = (const float*)d_in[1];   // 3*3
    const float* bias = (const float*)d_in[2];   // 1
    float* y = (float*)d_out;                    // 4094*4094

    dim3 grid((OUT_W + 127) / 128, (OUT_H + 15) / 16);  // (32, 256)
    conv3x3_wmma_f32<<<grid, 256, 0, stream>>>(x, wgt, bias, y);
}